// BilateralAffineTransform_31894427140448
// MI455X (gfx1250) — compile-verified
//
#include <hip/hip_runtime.h>

#define H_IMG 1080
#define W_IMG 1920
#define NPIX  (H_IMG * W_IMG)              // 2,073,600
#define GC 12
#define GL 8
#define GH 16
#define GW 16
#define GRID_ELEMS (GC * GL * GH * GW)     // 24576 floats = 96 KB
#define SLICE_STRIDE (GH * GW * GC + 4)    // 3076 dwords: +4 pad rotates banks per z
#define LDS_FLOATS (GL * SLICE_STRIDE)     // 24608 floats = 98,432 B

#define BLOCK_THREADS 256
#define NBLOCKS 1350
#define PIX_PER_THREAD 6                   // 1350*256*6 == NPIX exactly -> no tail guard
#define STAGE_ITERS (GRID_ELEMS / BLOCK_THREADS)   // 96, exact -> no divergence

#if defined(__gfx1250__) && __has_builtin(__builtin_amdgcn_global_load_async_to_lds_b32)
#define USE_ASYNC_LDS 1
#else
#define USE_ASYNC_LDS 0
#endif

typedef __attribute__((address_space(1))) int* gptr_i32;
typedef __attribute__((address_space(3))) int* lptr_i32;

__global__ void __launch_bounds__(BLOCK_THREADS, 1)
bilateral_slice_kernel(const float* __restrict__ rgb,
                       const float* __restrict__ grids,
                       const int*   __restrict__ grid_idx,
                       float*       __restrict__ out)
{
    __shared__ float sg[LDS_FLOATS];

    // ---- Stage the selected 96KB grid slice into LDS, transposed (c,z,y,x) -> (z,y,x,c).
    // Global side is perfectly coalesced (lane i reads dword i); the transpose happens
    // on the per-lane LDS scatter address via gfx1250 async-to-LDS. Uniform trip count
    // (96 per thread) -> no exec-mask loop guard, all lanes active for every async issue.
    const long gi = (long)grid_idx[0];
    const float* __restrict__ gbase = grids + gi * (long)GRID_ELEMS;

    for (int k = 0; k < STAGE_ITERS; ++k) {
        const int i   = k * BLOCK_THREADS + (int)threadIdx.x;
        const int c   = i >> 11;        // / (GL*GH*GW) = /2048
        const int rem = i & 2047;
        const int z   = rem >> 8;       // / (GH*GW)
        const int yx  = rem & 255;
        const int dst = z * SLICE_STRIDE + yx * GC + c;
#if USE_ASYNC_LDS
        __builtin_amdgcn_global_load_async_to_lds_b32(
            (gptr_i32)(gbase + i),
            (lptr_i32)(sg + dst),
            0, 0);
#else
        sg[dst] = gbase[i];
#endif
    }
#if USE_ASYNC_LDS
#if __has_builtin(__builtin_amdgcn_s_wait_asynccnt)
    __builtin_amdgcn_s_wait_asynccnt(0);
#else
    asm volatile("s_wait_asynccnt 0" ::: "memory");
#endif
#endif
    __syncthreads();

    // ---- Streaming pixel loop (HBM-bound: 24.9MB in + 99.5MB out).
    // Exact coverage: every thread handles PIX_PER_THREAD pixels, strided for coalescing.
    const int stride = NBLOCKS * BLOCK_THREADS;                 // 345,600
    const int tid0   = (int)(blockIdx.x * BLOCK_THREADS + threadIdx.x);

    for (int it = 0; it < PIX_PER_THREAD; ++it) {
        const int p   = tid0 + it * stride;
        const int row = p / W_IMG;
        const int col = p - row * W_IMG;

        const size_t rp = 3 * (size_t)p;
        // Prefetch the rgb stream for the next iteration (speculative; safe past end).
        __builtin_prefetch(rgb + rp + 3 * (size_t)stride, 0, 1);
        const float r = rgb[rp + 0];
        const float g = rgb[rp + 1];
        const float b = rgb[rp + 2];
        const float gray = fmaf(0.114f, b, fmaf(0.587f, g, 0.299f * r));

        // Match reference arithmetic: gx in [0,1], x = 2gx-1, ix = (x+1)*0.5*(Wg-1)
        const float gxv = (float)col * (1.0f / (float)(W_IMG - 1));
        const float gyv = (float)row * (1.0f / (float)(H_IMG - 1));
        const float fx = ((gxv * 2.0f - 1.0f) + 1.0f) * 0.5f * (float)(GW - 1);
        const float fy = ((gyv * 2.0f - 1.0f) + 1.0f) * 0.5f * (float)(GH - 1);
        const float zn = 2.0f * gray - 1.0f;
        const float fz = (zn + 1.0f) * 0.5f * (float)(GL - 1);

        const float x0f = fminf(fmaxf(floorf(fx), 0.0f), (float)(GW - 1));
        const float y0f = fminf(fmaxf(floorf(fy), 0.0f), (float)(GH - 1));
        const float z0f = fminf(fmaxf(floorf(fz), 0.0f), (float)(GL - 1));
        const float wx = fx - x0f, wy = fy - y0f, wz = fz - z0f;
        const int x0 = (int)x0f, y0 = (int)y0f, z0 = (int)z0f;
        const int x1 = min(x0 + 1, GW - 1);
        const int y1 = min(y0 + 1, GH - 1);
        const int z1 = min(z0 + 1, GL - 1);

        const int   zi[2]  = { z0, z1 };
        const int   yi[2]  = { y0, y1 };
        const int   xi[2]  = { x0, x1 };
        const float wzv[2] = { 1.0f - wz, wz };
        const float wyv[2] = { 1.0f - wy, wy };
        const float wxv[2] = { 1.0f - wx, wx };

        float acc[12];
#pragma unroll
        for (int k = 0; k < 12; ++k) acc[k] = 0.0f;

        // 8 corners, reference summation order (z outer, y, x inner).
#pragma unroll
        for (int a = 0; a < 2; ++a) {
#pragma unroll
            for (int bq = 0; bq < 2; ++bq) {
#pragma unroll
                for (int cq = 0; cq < 2; ++cq) {
                    const float w = (wzv[a] * wyv[bq]) * wxv[cq];
                    // channel-last cell: 12 contiguous floats, 16B aligned (48B stride)
                    const float4* cell = (const float4*)(sg + zi[a] * SLICE_STRIDE +
                                                         (yi[bq] * GW + xi[cq]) * GC);
                    const float4 v0 = cell[0];
                    const float4 v1 = cell[1];
                    const float4 v2 = cell[2];
                    acc[0]  = fmaf(v0.x, w, acc[0]);
                    acc[1]  = fmaf(v0.y, w, acc[1]);
                    acc[2]  = fmaf(v0.z, w, acc[2]);
                    acc[3]  = fmaf(v0.w, w, acc[3]);
                    acc[4]  = fmaf(v1.x, w, acc[4]);
                    acc[5]  = fmaf(v1.y, w, acc[5]);
                    acc[6]  = fmaf(v1.z, w, acc[6]);
                    acc[7]  = fmaf(v1.w, w, acc[7]);
                    acc[8]  = fmaf(v2.x, w, acc[8]);
                    acc[9]  = fmaf(v2.y, w, acc[9]);
                    acc[10] = fmaf(v2.z, w, acc[10]);
                    acc[11] = fmaf(v2.w, w, acc[11]);
                }
            }
        }

        // 48 bytes/pixel, 16B aligned -> three b128 stores.
        float4* op = (float4*)(out + (size_t)p * 12);
        op[0] = make_float4(acc[0], acc[1], acc[2],  acc[3]);
        op[1] = make_float4(acc[4], acc[5], acc[6],  acc[7]);
        op[2] = make_float4(acc[8], acc[9], acc[10], acc[11]);
    }
}

extern "C" void kernel_launch(void* const* d_in, const int* in_sizes, int n_in,
                              void* d_out, int out_size, void* d_ws, size_t ws_size,
                              hipStream_t stream) {
    (void)in_sizes; (void)n_in; (void)out_size; (void)d_ws; (void)ws_size;
    const float* rgb      = (const float*)d_in[0];
    const float* grids    = (const float*)d_in[1];
    const int*   grid_idx = (const int*)d_in[2];
    float*       out      = (float*)d_out;

    dim3 grid(NBLOCKS);
    dim3 block(BLOCK_THREADS);
    hipLaunchKernelGGL(bilateral_slice_kernel, grid, block, 0, stream,
                       rgb, grids, grid_idx, out);
}